// SparseLinear_43447889166980
// MI455X (gfx1250) — compile-verified
//
#include <hip/hip_runtime.h>
#include <hip/hip_bf16.h>

// SparseLinear shortlist scoring for MI455X (gfx1250, wave32).
// out[b,k] = dot(embed[b,:], sp_weight[shortlist[b,k],:]) + sp_bias[shortlist[b,k]]
// B=512, K=800, D=512, L=500000. Memory-bound gather-GEMV mapped onto
// V_WMMA_F32_16X16X4_F32 (fp32 matrix cores, exact reference precision).

typedef __attribute__((ext_vector_type(2))) float v2f;
typedef __attribute__((ext_vector_type(8))) float v8f;

#define BDIM 512
#define KDIM 800
#define DDIM 512
#define KTILES (KDIM / 16)          // 50 tiles of 16 shortlist rows
#define WAVES_PER_BLOCK 2

__global__ __launch_bounds__(32 * WAVES_PER_BLOCK)
void sparse_linear_wmma_kernel(const float* __restrict__ embed,      // [B, D]
                               const int*   __restrict__ shortlist,  // [B, K]
                               const float* __restrict__ sp_weight,  // [L, D]
                               const float* __restrict__ sp_bias,    // [L]
                               float*       __restrict__ out)        // [B, K]
{
    const int lane = threadIdx.x & 31;
    const int wave = threadIdx.x >> 5;
    const int wgid = blockIdx.x * WAVES_PER_BLOCK + wave;   // global wave id
    const int b    = wgid / KTILES;
    const int kt   = wgid % KTILES;

    const int* idx_base = shortlist + b * KDIM + kt * 16;

    // A-matrix fragment ownership (V_WMMA_F32_16X16X4_F32, wave32):
    //   lanes 0-15  : row M=lane,   v0=K0, v1=K1
    //   lanes 16-31 : row M=lane-16, v0=K2, v1=K3
    const int m    = lane & 15;
    const int half = (lane >> 4) & 1;           // 0 -> K{0,1}, 1 -> K{2,3}

    const long long row = (long long)idx_base[m];
    const float* wrow = sp_weight + row * (long long)DDIM + 2 * half;
    // B-matrix 4x16: every column = embed chunk; symmetric K placement.
    const float* erow = embed + (long long)b * DDIM + 2 * half;

    v8f acc0 = {};
    v8f acc1 = {};

    // March the K(=D) dimension in steps of 8 = two K=4 WMMAs on
    // independent accumulator chains (hides WMMA RAW latency).
    #pragma unroll 4
    for (int d = 0; d < DDIM; d += 8) {
        v2f a0 = *(const v2f*)(wrow + d);
        v2f b0 = *(const v2f*)(erow + d);
        v2f a1 = *(const v2f*)(wrow + d + 4);
        v2f b1 = *(const v2f*)(erow + d + 4);
        acc0 = __builtin_amdgcn_wmma_f32_16x16x4_f32(false, a0, false, b0,
                                                     (short)0, acc0, false, false);
        acc1 = __builtin_amdgcn_wmma_f32_16x16x4_f32(false, a1, false, b1,
                                                     (short)0, acc1, false, false);
    }

    float acc[8];
    #pragma unroll
    for (int i = 0; i < 8; ++i) acc[i] = acc0[i] + acc1[i];

    // D-matrix 16x16 f32 layout: column N=0 lives in lane 0 (rows 0-7 in
    // VGPR 0-7) and lane 16 (rows 8-15). Divergence only after all WMMAs.
    if ((lane & 15) == 0) {
        const int mbase = half * 8;
        float* o = out + (long long)b * KDIM + kt * 16 + mbase;
        #pragma unroll
        for (int r = 0; r < 8; ++r) {
            const int rr = idx_base[mbase + r];
            o[r] = acc[r] + sp_bias[rr];
        }
    }
}

extern "C" void kernel_launch(void* const* d_in, const int* in_sizes, int n_in,
                              void* d_out, int out_size, void* d_ws, size_t ws_size,
                              hipStream_t stream) {
    (void)in_sizes; (void)n_in; (void)out_size; (void)d_ws; (void)ws_size;

    const float* embed     = (const float*)d_in[0];   // [B, D]
    const int*   shortlist = (const int*)  d_in[1];   // [B, K]
    const float* sp_weight = (const float*)d_in[2];   // [L, D]
    const float* sp_bias   = (const float*)d_in[3];   // [L, 1] -> flat [L]
    float*       out       = (float*)d_out;           // [B, K]

    const int total_waves = BDIM * KTILES;                    // 25600
    const int blocks      = total_waves / WAVES_PER_BLOCK;    // 12800

    sparse_linear_wmma_kernel<<<blocks, 32 * WAVES_PER_BLOCK, 0, stream>>>(
        embed, shortlist, sp_weight, sp_bias, out);
}